// GAT_85134841741497
// MI455X (gfx1250) — compile-verified
//
#include <hip/hip_runtime.h>

#define D 128
#define NEG_SLOPE 0.2f

typedef __attribute__((ext_vector_type(2))) float v2f;
typedef __attribute__((ext_vector_type(8))) float v8f;

// ---------- helpers ----------

// Order-preserving map float -> uint so unsigned max == float max.
__device__ __forceinline__ unsigned orderF32(float f) {
    unsigned u = __float_as_uint(f);
    return (u & 0x80000000u) ? ~u : (u | 0x80000000u);
}
__device__ __forceinline__ float unorderF32(unsigned o) {
    unsigned u = (o & 0x80000000u) ? (o & 0x7FFFFFFFu) : ~o;
    return __uint_as_float(u);
}

__global__ void zero_u32(unsigned* __restrict__ p, long long n) {
    long long t = blockIdx.x * (long long)blockDim.x + threadIdx.x;
    if (t < n) p[t] = 0u;
}

// h[i,:] = emb[x[i],:]
__global__ void gather_rows(const float* __restrict__ emb, const int* __restrict__ x,
                            float* __restrict__ h, int N) {
    long long t = blockIdx.x * (long long)blockDim.x + threadIdx.x;
    if (t >= (long long)N * D) return;
    int row = (int)(t >> 7);
    int col = (int)(t & (D - 1));
    h[t] = emb[(size_t)x[row] * D + col];
}

// C[N,128] = A[N,128] @ W[128,128] (+ bias) using V_WMMA_F32_16X16X4_F32.
// One wave computes a 16x128 slab: 8 accumulator tiles of 16x16, K swept in
// steps of 4 (32 WMMAs per tile). W (64 KB) is staged once per workgroup into
// LDS in a pair-packed layout so each B fragment is one conflict-free
// ds_load_b64:
//   shw[((ka>>1)*128 + n)*2 + (ka&1)] = W[ka][n]
// A-frag (16x4 f32): lane m=lane&15 holds row m; hi=lane>>4 selects the K pair
// (k0+2*hi, k0+2*hi+1) -> contiguous v2f from the A row.
__global__ __launch_bounds__(256)
void gemm_wmma_f32(const float* __restrict__ A, const float* __restrict__ W,
                   const float* __restrict__ bias, float* __restrict__ C, int N) {
    __shared__ float shw[D * D];                 // 64 KB packed weight tile

    // Cooperative packed load of W (coalesced global read, packed LDS write).
    for (int i = threadIdx.x; i < D * D; i += 256) {
        int ka = i >> 7;                         // source row (K index)
        int n  = i & (D - 1);                    // source col (N index)
        shw[((((ka >> 1) * D) + n) << 1) | (ka & 1)] = W[i];
    }
    __syncthreads();

    const int lane = threadIdx.x & 31;
    const int wave = threadIdx.x >> 5;
    const int ml   = lane & 15;
    const int hi   = lane >> 4;
    const int row0 = (blockIdx.x * 8 + wave) * 16;
    if (row0 >= N) return;                       // after the barrier!

    v8f acc[8];
#pragma unroll
    for (int nt = 0; nt < 8; ++nt) {
        float bv = bias ? bias[nt * 16 + ml] : 0.0f;
#pragma unroll
        for (int r = 0; r < 8; ++r) acc[nt][r] = bv;
    }

    const float* arow = A + (size_t)(row0 + ml) * D;
    for (int k0 = 0; k0 < D; k0 += 4) {
        const int ka = k0 + 2 * hi;
        v2f a = *(const v2f*)(arow + ka);        // A[row0+ml][ka..ka+1]
        const float* bp = shw + (size_t)(ka >> 1) * D * 2;
#pragma unroll
        for (int nt = 0; nt < 8; ++nt) {
            v2f b = *(const v2f*)(bp + ((nt * 16 + ml) << 1));  // ds_load_b64
            acc[nt] = __builtin_amdgcn_wmma_f32_16x16x4_f32(
                false, a, false, b, (short)0, acc[nt], false, false);
        }
    }

    // C/D layout: VGPR r -> row (r + 8*hi), col = ml within each 16-col tile.
#pragma unroll
    for (int nt = 0; nt < 8; ++nt)
#pragma unroll
        for (int r = 0; r < 8; ++r)
            C[(size_t)(row0 + r + 8 * hi) * D + nt * 16 + ml] = acc[nt][r];
}

// es[i] = dot(g[i,:], a_s); ed[i] = dot(g[i,:], a_d). One wave32 per node.
__global__ void attn_coef(const float* __restrict__ g, const float* __restrict__ a_s,
                          const float* __restrict__ a_d, float* __restrict__ es,
                          float* __restrict__ ed, int N) {
    long long node = (blockIdx.x * (long long)blockDim.x + threadIdx.x) >> 5;
    int lane = threadIdx.x & 31;
    if (node >= N) return;
    const float4 gv = *(const float4*)(g + (size_t)node * D + lane * 4);
    const float4 sv = *(const float4*)(a_s + lane * 4);
    const float4 dv = *(const float4*)(a_d + lane * 4);
    float ps = gv.x * sv.x + gv.y * sv.y + gv.z * sv.z + gv.w * sv.w;
    float pd = gv.x * dv.x + gv.y * dv.y + gv.z * dv.z + gv.w * dv.w;
#pragma unroll
    for (int off = 16; off > 0; off >>= 1) {
        ps += __shfl_xor(ps, off, 32);
        pd += __shfl_xor(pd, off, 32);
    }
    if (lane == 0) { es[node] = ps; ed[node] = pd; }
}

// e[j] = leakyrelu(es[src]+ed[dst]); m[dst] = max(m[dst], e[j])  (ordered-uint atomic max)
__global__ void edge_score_max(const int* __restrict__ src, const int* __restrict__ dst,
                               const float* __restrict__ es, const float* __restrict__ ed,
                               float* __restrict__ e, unsigned* __restrict__ m,
                               long long E, long long Etot) {
    long long j = blockIdx.x * (long long)blockDim.x + threadIdx.x;
    if (j >= Etot) return;
    int s = (j < E) ? src[j] : (int)(j - E);     // implicit self-loops for j >= E
    int d = (j < E) ? dst[j] : (int)(j - E);
    float v = es[s] + ed[d];
    v = (v >= 0.0f) ? v : NEG_SLOPE * v;
    e[j] = v;
    __hip_atomic_fetch_max(m + d, orderF32(v), __ATOMIC_RELAXED, __HIP_MEMORY_SCOPE_AGENT);
}

// w[j] = exp(e[j]-m[dst]); z[dst] += w[j]  (hardware f32 atomic add)
__global__ void edge_exp_sum(const int* __restrict__ dst, float* __restrict__ e,
                             const unsigned* __restrict__ m, float* __restrict__ z,
                             long long E, long long Etot) {
    long long j = blockIdx.x * (long long)blockDim.x + threadIdx.x;
    if (j >= Etot) return;
    int d = (j < E) ? dst[j] : (int)(j - E);
    float wv = expf(e[j] - unorderF32(m[d]));
    e[j] = wv;
    __hip_atomic_fetch_add(z + d, wv, __ATOMIC_RELAXED, __HIP_MEMORY_SCOPE_AGENT);
}

// out[dst,:] += (w[j]/z[dst]) * g[src,:]. One wave32 per edge, float4 per lane.
__global__ void edge_aggregate(const int* __restrict__ src, const int* __restrict__ dst,
                               const float* __restrict__ g, const float* __restrict__ w,
                               const float* __restrict__ z, float* __restrict__ out,
                               long long E, long long Etot) {
    long long j = (blockIdx.x * (long long)blockDim.x + threadIdx.x) >> 5;
    int lane = threadIdx.x & 31;
    if (j >= Etot) return;
    int s = (j < E) ? src[j] : (int)(j - E);
    int d = (j < E) ? dst[j] : (int)(j - E);
    float alpha = w[j] / z[d];
    const float4 gv = *(const float4*)(g + (size_t)s * D + lane * 4);
    float* o = out + (size_t)d * D + lane * 4;
    __hip_atomic_fetch_add(o + 0, alpha * gv.x, __ATOMIC_RELAXED, __HIP_MEMORY_SCOPE_AGENT);
    __hip_atomic_fetch_add(o + 1, alpha * gv.y, __ATOMIC_RELAXED, __HIP_MEMORY_SCOPE_AGENT);
    __hip_atomic_fetch_add(o + 2, alpha * gv.z, __ATOMIC_RELAXED, __HIP_MEMORY_SCOPE_AGENT);
    __hip_atomic_fetch_add(o + 3, alpha * gv.w, __ATOMIC_RELAXED, __HIP_MEMORY_SCOPE_AGENT);
}

// h[i,c] = prelu(agg[i,c] + b[c], p)
__global__ void bias_prelu(const float* __restrict__ in, const float* __restrict__ b,
                           const float* __restrict__ p, float* __restrict__ out, int N) {
    long long t = blockIdx.x * (long long)blockDim.x + threadIdx.x;
    if (t >= (long long)N * D) return;
    int col = (int)(t & (D - 1));
    float v = in[t] + b[col];
    out[t] = (v >= 0.0f) ? v : p[0] * v;
}

// ---------- launch ----------

extern "C" void kernel_launch(void* const* d_in, const int* in_sizes, int n_in,
                              void* d_out, int out_size, void* d_ws, size_t ws_size,
                              hipStream_t stream) {
    (void)n_in; (void)out_size; (void)ws_size;

    const int*   x   = (const int*)d_in[0];
    const int*   ei  = (const int*)d_in[1];
    // d_in[2] = edge_weight: unused by the reference.
    const float* emb = (const float*)d_in[3];
    const float* Wl[3]  = {(const float*)d_in[4],  (const float*)d_in[9],  (const float*)d_in[14]};
    const float* asl[3] = {(const float*)d_in[5],  (const float*)d_in[10], (const float*)d_in[15]};
    const float* adl[3] = {(const float*)d_in[6],  (const float*)d_in[11], (const float*)d_in[16]};
    const float* bl[3]  = {(const float*)d_in[7],  (const float*)d_in[12], (const float*)d_in[17]};
    const float* pl[3]  = {(const float*)d_in[8],  (const float*)d_in[13], (const float*)d_in[18]};
    const float* Wo = (const float*)d_in[19];
    const float* bo = (const float*)d_in[20];

    const int       N    = in_sizes[0];
    const long long E    = in_sizes[1] / 2;
    const long long Etot = E + N;
    const int* srcIdx = ei;        // edge_index[0,:]
    const int* dstIdx = ei + E;    // edge_index[1,:]

    // Workspace layout (all 256B aligned; agg|z|m contiguous for one-shot zeroing).
    char* wsp = (char*)d_ws;
    auto carve = [&](size_t bytes) {
        char* r = wsp;
        wsp += (bytes + 255) & ~(size_t)255;
        return r;
    };
    const size_t NDb = (size_t)N * D * sizeof(float);
    float*    h   = (float*)carve(NDb);
    float*    g   = (float*)carve(NDb);
    float*    agg = (float*)carve(NDb + 2 * (size_t)N * sizeof(float));
    float*    z   = agg + (size_t)N * D;
    unsigned* m   = (unsigned*)(z + N);
    float*    es  = (float*)carve((size_t)N * sizeof(float));
    float*    ed  = (float*)carve((size_t)N * sizeof(float));
    float*    ew  = (float*)carve((size_t)Etot * sizeof(float));

    const dim3 blk(256);
    const long long ndThreads = (long long)N * D;
    const int ndBlocks       = (int)((ndThreads + 255) / 256);
    const int gemmBlocks     = ((N + 15) / 16 + 7) / 8;        // 8 waves/WG, 16 rows/wave
    const int nodeWaveBlocks = (N + 7) / 8;                    // 1 wave per node
    const int edgeBlocks     = (int)((Etot + 255) / 256);      // 1 thread per edge
    const int edgeWaveBlocks = (int)((Etot + 7) / 8);          // 1 wave per edge

    gather_rows<<<ndBlocks, blk, 0, stream>>>(emb, x, h, N);

    for (int l = 0; l < 3; ++l) {
        const long long zn = (long long)N * D + 2LL * N;       // agg + z + m
        zero_u32<<<(int)((zn + 255) / 256), blk, 0, stream>>>((unsigned*)agg, zn);
        gemm_wmma_f32<<<gemmBlocks, blk, 0, stream>>>(h, Wl[l], nullptr, g, N);
        attn_coef<<<nodeWaveBlocks, blk, 0, stream>>>(g, asl[l], adl[l], es, ed, N);
        edge_score_max<<<edgeBlocks, blk, 0, stream>>>(srcIdx, dstIdx, es, ed, ew, m, E, Etot);
        edge_exp_sum<<<edgeBlocks, blk, 0, stream>>>(dstIdx, ew, m, z, E, Etot);
        edge_aggregate<<<edgeWaveBlocks, blk, 0, stream>>>(srcIdx, dstIdx, g, ew, z, agg, E, Etot);
        bias_prelu<<<ndBlocks, blk, 0, stream>>>(agg, bl[l], pl[l], h, N);
    }

    gemm_wmma_f32<<<gemmBlocks, blk, 0, stream>>>(h, Wo, bo, (float*)d_out, N);
}